// TopKSAE_48189533061420
// MI455X (gfx1250) — compile-verified
//
#include <hip/hip_runtime.h>

typedef float v2f __attribute__((ext_vector_type(2)));
typedef float v8f __attribute__((ext_vector_type(8)));
typedef int   v4i __attribute__((ext_vector_type(4)));

#define INPUT_DIM  1024
#define LATENT_DIM 16384
#define KTOP       64
#define BATCH      8192

// ---------------- CDNA5 async copy global->LDS helpers ----------------
typedef __attribute__((address_space(1))) v4i Gv4i;   // global int4
typedef __attribute__((address_space(3))) v4i Lv4i;   // LDS int4

__device__ __forceinline__ void async_ld16(const void* g, void* l) {
#if defined(__gfx1250__) && __has_builtin(__builtin_amdgcn_global_load_async_to_lds_b128)
    // GLOBAL_LOAD_ASYNC_TO_LDS_B128: per-lane 16B global->LDS, tracked by ASYNCcnt
    __builtin_amdgcn_global_load_async_to_lds_b128(
        (Gv4i*)(void*)g, (Lv4i*)l, 0, 0);
#else
    *(float4*)l = *(const float4*)g;   // sync fallback: global->VGPR->ds_store
#endif
}

__device__ __forceinline__ void wait_async0() {
#if defined(__gfx1250__)
#  if __has_builtin(__builtin_amdgcn_s_wait_asynccnt)
    __builtin_amdgcn_s_wait_asynccnt(0);
#  else
    asm volatile("s_wait_asynccnt 0" ::: "memory");
#  endif
#endif
}

// ---------------- Encoder: z = relu(x @ W_enc^T + b_enc) ----------------
// C tile 64(M) x 128(N) per 256-thread workgroup (8 waves, wave32).
// Each wave: 32x32 subtile = 2x2 WMMA 16x16 f32 accumulators.
// Double-buffered LDS staged with async global->LDS copies.
#define BM 64
#define BN 128
#define BK 32
#define LDS_STRIDE (BK + 1)   // 33 floats: odd stride -> conflict-free column reads

__global__ __launch_bounds__(256)
void sae_encoder(const float* __restrict__ x,
                 const float* __restrict__ W_enc,
                 const float* __restrict__ b_enc,
                 float* __restrict__ z_out)
{
    __shared__ float xs [2][BM][LDS_STRIDE];   // 2 x  64 x 33 floats
    __shared__ float wsd[2][BN][LDS_STRIDE];   // 2 x 128 x 33 floats (~50 KB total)

    const int tid  = threadIdx.x;
    const int wave = tid >> 5;
    const int lane = tid & 31;
    const int m_base = blockIdx.y * BM;
    const int n_base = blockIdx.x * BN;
    const int wr = wave >> 2;      // 0..1 : wave M offset (x32)
    const int wc = wave & 3;       // 0..3 : wave N offset (x32)

    const int lrow = lane & 15;          // M (for A) / N (for B) within 16
    const int koff = (lane >> 4) * 2;    // K pair selected by lane half

    v8f acc[2][2];
    #pragma unroll
    for (int mt = 0; mt < 2; ++mt)
        #pragma unroll
        for (int nt = 0; nt < 2; ++nt)
            acc[mt][nt] = (v8f){0.f,0.f,0.f,0.f,0.f,0.f,0.f,0.f};

    const int lr = tid >> 3;            // 0..31
    const int lc = (tid & 7) * 4;       // 0..28 step 4 (16B chunk within one row)

    auto stage = [&](int buf, int k0) {
        #pragma unroll
        for (int p = 0; p < 2; ++p) {
            const int row = lr + p * 32;
            async_ld16(x + (size_t)(m_base + row) * INPUT_DIM + k0 + lc,
                       &xs[buf][row][lc]);
        }
        #pragma unroll
        for (int p = 0; p < 4; ++p) {
            const int row = lr + p * 32;
            async_ld16(W_enc + (size_t)(n_base + row) * INPUT_DIM + k0 + lc,
                       &wsd[buf][row][lc]);
        }
    };

    stage(0, 0);
    int cur = 0;

    for (int k0 = 0; k0 < INPUT_DIM; k0 += BK) {
        wait_async0();          // my async copies into buf[cur] have landed
        __syncthreads();        // everyone's have landed; prev reads of other buf done
        if (k0 + BK < INPUT_DIM)
            stage(cur ^ 1, k0 + BK);   // prefetch next stage while computing

        #pragma unroll
        for (int kk = 0; kk < BK; kk += 4) {
            v2f a[2], b[2];
            #pragma unroll
            for (int mt = 0; mt < 2; ++mt) {
                const int m = wr * 32 + mt * 16 + lrow;
                a[mt].x = xs[cur][m][kk + koff];
                a[mt].y = xs[cur][m][kk + koff + 1];
            }
            #pragma unroll
            for (int nt = 0; nt < 2; ++nt) {
                const int n = wc * 32 + nt * 16 + lrow;
                b[nt].x = wsd[cur][n][kk + koff];
                b[nt].y = wsd[cur][n][kk + koff + 1];
            }
            #pragma unroll
            for (int mt = 0; mt < 2; ++mt)
                #pragma unroll
                for (int nt = 0; nt < 2; ++nt)
                    acc[mt][nt] = __builtin_amdgcn_wmma_f32_16x16x4_f32(
                        false, a[mt], false, b[nt],
                        (short)0, acc[mt][nt], false, false);
        }
        cur ^= 1;
    }

    // epilogue: bias + relu + store z
    #pragma unroll
    for (int mt = 0; mt < 2; ++mt) {
        #pragma unroll
        for (int nt = 0; nt < 2; ++nt) {
            const int n = n_base + wc * 32 + nt * 16 + (lane & 15);
            const float bias = b_enc[n];
            #pragma unroll
            for (int i = 0; i < 8; ++i) {
                const int m = m_base + wr * 32 + mt * 16 + i + 8 * (lane >> 4);
                float v = acc[mt][nt][i] + bias;
                v = v > 0.f ? v : 0.f;
                z_out[(size_t)m * LATENT_DIM + n] = v;
            }
        }
    }
}

// ---------------- TopK: exact 64th-largest via 4-pass byte radix ----------------
// One block per row. 64 values/thread in registers (relu => bits monotonic).
__global__ __launch_bounds__(256)
void sae_topk(float* __restrict__ z,
              int*   __restrict__ ws_idx,
              float* __restrict__ ws_val)
{
    const int row = blockIdx.x;
    const int tid = threadIdx.x;
    float* zrow = z + (size_t)row * LATENT_DIM;

    unsigned int u[64];
    #pragma unroll
    for (int j = 0; j < 64; ++j)
        u[j] = __float_as_uint(zrow[j * 256 + tid]);

    __shared__ int      hist[256];
    __shared__ unsigned spref;
    __shared__ int      sremk;
    __shared__ int      s_gt_cnt, s_eq_cnt;
    if (tid == 0) { spref = 0u; sremk = KTOP; }
    __syncthreads();

    for (int bp = 3; bp >= 0; --bp) {
        hist[tid] = 0;
        __syncthreads();
        const unsigned pref  = spref;
        const unsigned pmask = (bp == 3) ? 0u : (0xFFFFFFFFu << ((bp + 1) * 8));
        #pragma unroll
        for (int j = 0; j < 64; ++j) {
            if ((u[j] & pmask) == pref)
                atomicAdd(&hist[(u[j] >> (bp * 8)) & 255], 1);
        }
        __syncthreads();
        if (tid == 0) {
            int remk = sremk, cum = 0, b = 255;
            for (; b > 0; --b) {
                if (cum + hist[b] >= remk) break;
                cum += hist[b];
            }
            sremk = remk - cum;                       // # of ==T still needed
            spref = spref | ((unsigned)b << (bp * 8));
        }
        __syncthreads();
    }

    const unsigned T      = spref;         // bit pattern of 64th-largest value
    const int      needEq = sremk;         // ties at T to accept
    const int      gtTot  = KTOP - needEq; // elements strictly > T
    if (tid == 0) { s_gt_cnt = 0; s_eq_cnt = 0; }
    __syncthreads();

    #pragma unroll
    for (int j = 0; j < 64; ++j) {
        const int idx = j * 256 + tid;
        const unsigned b = u[j];
        const float v = __uint_as_float(b);
        bool keep = false;
        if (b > T) {
            keep = true;
            const int slot = atomicAdd(&s_gt_cnt, 1);
            ws_idx[(size_t)row * KTOP + slot] = idx;
            ws_val[(size_t)row * KTOP + slot] = v;
        } else if (b == T) {
            const int e = atomicAdd(&s_eq_cnt, 1);
            if (e < needEq) {
                keep = true;
                const int slot = gtTot + e;
                ws_idx[(size_t)row * KTOP + slot] = idx;
                ws_val[(size_t)row * KTOP + slot] = v;
            }
        }
        zrow[idx] = keep ? v : 0.0f;
    }
}

// ---------------- Decoder: recon = z_sparse @ W_dec^T + b_dec ----------------
// One block per row; 64 gathered columns of W_dec (L2-resident: 64MB < 192MB L2).
__global__ __launch_bounds__(256)
void sae_decoder(const float* __restrict__ W_dec,
                 const float* __restrict__ b_dec,
                 const int*   __restrict__ ws_idx,
                 const float* __restrict__ ws_val,
                 float* __restrict__ recon)
{
    const int row = blockIdx.x;
    const int tid = threadIdx.x;
    __shared__ int   sidx[KTOP];
    __shared__ float sval[KTOP];
    if (tid < KTOP) {
        sidx[tid] = ws_idx[(size_t)row * KTOP + tid];
        sval[tid] = ws_val[(size_t)row * KTOP + tid];
    }
    __syncthreads();

    float acc[4];
    #pragma unroll
    for (int q = 0; q < 4; ++q) acc[q] = b_dec[tid + q * 256];

    for (int j = 0; j < KTOP; ++j) {
        const float v = sval[j];
        const int   l = sidx[j];
        #pragma unroll
        for (int q = 0; q < 4; ++q)
            acc[q] = fmaf(v, W_dec[(size_t)(tid + q * 256) * LATENT_DIM + l], acc[q]);
    }
    #pragma unroll
    for (int q = 0; q < 4; ++q)
        recon[(size_t)row * INPUT_DIM + tid + q * 256] = acc[q];
}

// ---------------- launch ----------------
extern "C" void kernel_launch(void* const* d_in, const int* in_sizes, int n_in,
                              void* d_out, int out_size, void* d_ws, size_t ws_size,
                              hipStream_t stream)
{
    const float* x     = (const float*)d_in[0];
    const float* W_enc = (const float*)d_in[1];
    const float* b_enc = (const float*)d_in[2];
    const float* W_dec = (const float*)d_in[3];
    const float* b_dec = (const float*)d_in[4];

    float* recon = (float*)d_out;                                   // [8192,1024]
    float* z     = (float*)d_out + (size_t)BATCH * INPUT_DIM;       // [8192,16384]

    int*   ws_idx = (int*)d_ws;                                     // 8192*64 ints
    float* ws_val = (float*)((char*)d_ws + (size_t)BATCH * KTOP * sizeof(int));

    dim3 egrid(LATENT_DIM / BN, BATCH / BM);   // 128 x 128 workgroups
    sae_encoder<<<egrid, 256, 0, stream>>>(x, W_enc, b_enc, z);
    sae_topk   <<<BATCH, 256, 0, stream>>>(z, ws_idx, ws_val);
    sae_decoder<<<BATCH, 256, 0, stream>>>(W_dec, b_dec, ws_idx, ws_val, recon);
}